// EC_CA_SA2_90082644067001
// MI455X (gfx1250) — compile-verified
//
#include <hip/hip_runtime.h>
#include <math.h>

typedef __attribute__((ext_vector_type(2))) float v2f;
typedef __attribute__((ext_vector_type(8))) float v8f;

#define NB  32      // batch
#define NC  256     // channels
#define NH  64
#define NW  64
#define NM  16      // MCH
#define NL  128     // 2*H
#define HW  4096    // H*W
#define EPSV 1e-5f

// ---- workspace layout (float offsets) ----
#define OFF_CAT   0u          // (B, C, 128)        1048576
#define OFF_Y     1048576u    // (B, 16, 128)         65536
#define OFF_S     1114112u    // (B, C, 128)        1048576
#define OFF_POOL  2162688u    // (B, C)                8192
#define OFF_FCH   2170880u    // (B, C)                8192
#define OFF_CMP   2179072u    // (8, B, 4096)       1048576
#define OFF_CM    3227648u    // (B, 4096)           131072
#define OFF_FSA   3358720u    // (B, 4096)           131072
// total: 3489792 floats = ~13.3 MB of d_ws

// ---------------------------------------------------------------------------
// Pass 1: one read of x produces x_h, x_w (into cat), pooled, and per-chunk
// channel-mean partials. One block per (b, 32-channel chunk).
// ---------------------------------------------------------------------------
__global__ void k_reduce(const float* __restrict__ x,
                         float* __restrict__ cat,
                         float* __restrict__ pooled,
                         float* __restrict__ cmean_part) {
  const int cc = blockIdx.x;           // 0..7 channel chunk
  const int b  = blockIdx.y;           // 0..31
  const int t  = threadIdx.x;          // 0..255

  __shared__ float lds[64 * 65];       // padded plane (bank-conflict free)
  __shared__ float srow[64];

  float4 accm[4];                      // channel-mean partial accumulators
  accm[0] = accm[1] = accm[2] = accm[3] = make_float4(0.f, 0.f, 0.f, 0.f);

  for (int ci = 0; ci < 32; ++ci) {
    const int c = cc * 32 + ci;
    const float4* p4 =
        (const float4*)(x + (((size_t)(b * NC + c)) << 12));

    float4 v[4];
#pragma unroll
    for (int j = 0; j < 4; ++j) {
      const int i = t + 256 * j;
      v[j] = p4[i];
      accm[j].x += v[j].x; accm[j].y += v[j].y;
      accm[j].z += v[j].z; accm[j].w += v[j].w;
      const int row = i >> 4;               // 16 float4 per row
      const int col = (i & 15) << 2;
      float* dst = &lds[row * 65 + col];
      dst[0] = v[j].x; dst[1] = v[j].y; dst[2] = v[j].z; dst[3] = v[j].w;
    }
    __syncthreads();

    if (t < 64) {                           // row sums -> x_h
      float rs = 0.f;
      const float* rp = &lds[t * 65];
#pragma unroll
      for (int w = 0; w < 64; ++w) rs += rp[w];
      cat[((size_t)(b * NC + c)) * NL + t] = rs * (1.f / 64.f);
      srow[t] = rs;
    } else if (t < 128) {                   // column sums -> x_w
      const int w = t - 64;
      float cs = 0.f;
#pragma unroll
      for (int h = 0; h < 64; ++h) cs += lds[h * 65 + w];
      cat[((size_t)(b * NC + c)) * NL + 64 + w] = cs * (1.f / 64.f);
    }
    __syncthreads();

    if (t == 0) {
      float tot = 0.f;
#pragma unroll
      for (int h = 0; h < 64; ++h) tot += srow[h];
      pooled[b * NC + c] = tot * (1.f / 4096.f);
    }
  }

  // write scaled channel-mean partial for this chunk: layout (cc, b, 4096)
  float4* cp4 = (float4*)(cmean_part + (((size_t)(cc * NB + b)) << 12));
#pragma unroll
  for (int j = 0; j < 4; ++j) {
    const int i = t + 256 * j;
    float4 o = accm[j];
    o.x *= (1.f / 256.f); o.y *= (1.f / 256.f);
    o.z *= (1.f / 256.f); o.w *= (1.f / 256.f);
    cp4[i] = o;
  }
}

// deterministic channel-mean: fixed-order sum of the 8 chunk partials
__global__ void k_cmean_sum(const float* __restrict__ part,
                            float* __restrict__ cmean) {
  const int idx = blockIdx.x * 256 + threadIdx.x;   // 131072
  const int b = idx >> 12, p = idx & 4095;
  float s = 0.f;
#pragma unroll
  for (int cc = 0; cc < 8; ++cc) s += part[(((size_t)(cc * NB + b)) << 12) + p];
  cmean[idx] = s;
}

// f_sa = sigmoid(conv3x3(channel-mean)), SAME zero padding
__global__ void k_fsa(const float* __restrict__ cmean,
                      const float* __restrict__ w21,
                      float* __restrict__ fsa) {
  const int idx = blockIdx.x * 256 + threadIdx.x;   // 131072
  const int b = idx >> 12, r = idx & 4095;
  const int h = r >> 6, w = r & 63;
  float acc = 0.f;
#pragma unroll
  for (int i = 0; i < 3; ++i)
#pragma unroll
    for (int j = 0; j < 3; ++j) {
      const int hh = h + i - 1, ww = w + j - 1;
      if (hh >= 0 && hh < 64 && ww >= 0 && ww < 64)
        acc += cmean[(b << 12) + hh * 64 + ww] * w21[i * 3 + j];
    }
  fsa[idx] = 1.f / (1.f + expf(-acc));
}

// f_ch = conv1d(pooled, w20, K=5, SAME) along the channel axis
__global__ void k_fch(const float* __restrict__ pooled,
                      const float* __restrict__ w20,
                      float* __restrict__ fch) {
  const int idx = blockIdx.x * 256 + threadIdx.x;   // 8192
  const int b = idx >> 8, c = idx & 255;
  float acc = 0.f;
#pragma unroll
  for (int k = 0; k < 5; ++k) {
    const int cs = c + k - 2;
    if (cs >= 0 && cs < NC) acc += pooled[b * NC + cs] * w20[k];
  }
  fch[idx] = acc;
}

// ---------------------------------------------------------------------------
// GEMM1: y[b] (16x128) = relu(bn(w10 (16x256) @ cat[b] (256x128)))
// one block per b; 8 waves; wave = 16-wide N tile; V_WMMA_F32_16X16X4_F32
// ---------------------------------------------------------------------------
__global__ void k_gemm1(const float* __restrict__ w10,
                        const float* __restrict__ cat,
                        const float* __restrict__ gamma,
                        const float* __restrict__ beta,
                        const float* __restrict__ mean,
                        const float* __restrict__ var,
                        float* __restrict__ y) {
  const int b    = blockIdx.x;
  const int wave = threadIdx.x >> 5;
  const int lane = threadIdx.x & 31;

  __shared__ float sinv[16], sbias[16];
  if (threadIdx.x < 16) {
    const float iv = rsqrtf(var[threadIdx.x] + EPSV) * gamma[threadIdx.x];
    sinv[threadIdx.x]  = iv;
    sbias[threadIdx.x] = beta[threadIdx.x] - mean[threadIdx.x] * iv;
  }
  __syncthreads();

  const float* catb = cat + (size_t)b * NC * NL;
  const int n      = (wave << 4) + (lane & 15);  // output column
  const int m      = lane & 15;                  // A row for this lane
  const int khalf  = (lane >> 4) << 1;           // lanes 16..31 hold K=2,3

  v8f acc = {0.f, 0.f, 0.f, 0.f, 0.f, 0.f, 0.f, 0.f};
  for (int k0 = 0; k0 < NC; k0 += 4) {
    v2f a, bb;
    a.x  = w10[m * NC + k0 + khalf];
    a.y  = w10[m * NC + k0 + khalf + 1];
    bb.x = catb[(size_t)(k0 + khalf) * NL + n];
    bb.y = catb[(size_t)(k0 + khalf + 1) * NL + n];
    acc = __builtin_amdgcn_wmma_f32_16x16x4_f32(false, a, false, bb,
                                                (short)0, acc, false, false);
  }

  const int mbase = (lane >> 4) << 3;            // D: M = r + 8*(lane>=16)
  float* yb = y + (size_t)b * NM * NL;
#pragma unroll
  for (int r = 0; r < 8; ++r) {
    const int mm = r + mbase;
    float v = acc[r] * sinv[mm] + sbias[mm];
    yb[mm * NL + n] = fmaxf(v, 0.f);
  }
}

// ---------------------------------------------------------------------------
// GEMM2: s[b] (256x128) = sigmoid(w11 (256x16) @ y[b] (16x128))
// one block per (b, 16-row M tile); 8 waves cover the 8 N tiles
// ---------------------------------------------------------------------------
__global__ void k_gemm2(const float* __restrict__ w11,
                        const float* __restrict__ y,
                        float* __restrict__ s) {
  const int bm   = blockIdx.x;            // b*16 + mt
  const int b    = bm >> 4;
  const int mt   = bm & 15;
  const int wave = threadIdx.x >> 5;
  const int lane = threadIdx.x & 31;

  const float* yb  = y + (size_t)b * NM * NL;
  const int n      = (wave << 4) + (lane & 15);
  const int row    = mt * 16 + (lane & 15);      // A row (channel c)
  const int khalf  = (lane >> 4) << 1;

  v8f acc = {0.f, 0.f, 0.f, 0.f, 0.f, 0.f, 0.f, 0.f};
#pragma unroll
  for (int k0 = 0; k0 < NM; k0 += 4) {
    v2f a, bb;
    a.x  = w11[row * NM + k0 + khalf];
    a.y  = w11[row * NM + k0 + khalf + 1];
    bb.x = yb[(k0 + khalf) * NL + n];
    bb.y = yb[(k0 + khalf + 1) * NL + n];
    acc = __builtin_amdgcn_wmma_f32_16x16x4_f32(false, a, false, bb,
                                                (short)0, acc, false, false);
  }

  float* sb = s + (size_t)b * NC * NL;
  const int cbase = mt * 16 + ((lane >> 4) << 3);
#pragma unroll
  for (int r = 0; r < 8; ++r) {
    const int c = cbase + r;
    sb[(size_t)c * NL + n] = 1.f / (1.f + expf(-acc[r]));
  }
}

// ---------------------------------------------------------------------------
// Final: out = x * (f_ch[b,c] + s_h[b,c,h]*s_w[b,c,w] + f_sa[b,h,w])
// one block per (b,c) plane, float4 throughout
// ---------------------------------------------------------------------------
__global__ void k_final(const float* __restrict__ x,
                        const float* __restrict__ s,
                        const float* __restrict__ fch,
                        const float* __restrict__ fsa,
                        float* __restrict__ out) {
  const int bc = blockIdx.x;                    // b*256 + c
  const int b  = bc >> 8;
  const int t  = threadIdx.x;

  const float  fc    = fch[bc];
  const float* srow  = s + (size_t)bc * NL;     // [0:64]=s_h, [64:128]=s_w
  const float* fsab  = fsa + ((size_t)b << 12);
  const float4* x4   = (const float4*)(x + ((size_t)bc << 12));
  float4*       o4   = (float4*)(out + ((size_t)bc << 12));
  const float4* sw4  = (const float4*)(srow + 64);
  const float4* fs4  = (const float4*)fsab;

#pragma unroll
  for (int j = 0; j < 4; ++j) {
    const int i  = t + 256 * j;                 // float4 index in plane
    const int h  = i >> 4;                      // 16 float4 per row
    const int q  = i & 15;                      // float4 col
    const float  sh = srow[h];
    const float4 xv = x4[i];
    const float4 sw = sw4[q];
    const float4 fv = fs4[(h << 4) + q];
    float4 ov;
    ov.x = xv.x * (fc + sh * sw.x + fv.x);
    ov.y = xv.y * (fc + sh * sw.y + fv.y);
    ov.z = xv.z * (fc + sh * sw.z + fv.z);
    ov.w = xv.w * (fc + sh * sw.w + fv.w);
    o4[i] = ov;
  }
}

// ---------------------------------------------------------------------------
extern "C" void kernel_launch(void* const* d_in, const int* in_sizes, int n_in,
                              void* d_out, int out_size, void* d_ws, size_t ws_size,
                              hipStream_t stream) {
  const float* x     = (const float*)d_in[0];
  const float* w10   = (const float*)d_in[1];
  const float* w11   = (const float*)d_in[2];
  const float* gamma = (const float*)d_in[3];
  const float* beta  = (const float*)d_in[4];
  const float* mean  = (const float*)d_in[5];
  const float* var   = (const float*)d_in[6];
  const float* w20   = (const float*)d_in[7];
  const float* w21   = (const float*)d_in[8];
  float* out = (float*)d_out;

  float* ws    = (float*)d_ws;
  float* cat   = ws + OFF_CAT;
  float* y     = ws + OFF_Y;
  float* s     = ws + OFF_S;
  float* pool  = ws + OFF_POOL;
  float* fch   = ws + OFF_FCH;
  float* cmp   = ws + OFF_CMP;
  float* cm    = ws + OFF_CM;
  float* fsa   = ws + OFF_FSA;

  k_reduce   <<<dim3(8, NB), 256, 0, stream>>>(x, cat, pool, cmp);
  k_cmean_sum<<<512,  256, 0, stream>>>(cmp, cm);
  k_fsa      <<<512,  256, 0, stream>>>(cm, w21, fsa);
  k_fch      <<<32,   256, 0, stream>>>(pool, w20, fch);
  k_gemm1    <<<NB,   256, 0, stream>>>(w10, cat, gamma, beta, mean, var, y);
  k_gemm2    <<<NB * 16, 256, 0, stream>>>(w11, y, s);
  k_final    <<<NB * NC, 256, 0, stream>>>(x, s, fch, fsa, out);
}